// RGB2Mixed_49555332661392
// MI455X (gfx1250) — compile-verified
//
#include <hip/hip_runtime.h>

// Elementwise shift-subtract + copy:
//   out[:, 0:3]  = x - roll(x, 1, axis=w)
//   out[:, 3:6]  = x - roll(x, 1, axis=h)
//   out[:, 6:9]  = x
// Memory-bound: ~805 MB total traffic -> ~34.6 us floor at 23.3 TB/s.
// Strategy: async-stage rows into LDS (CDNA5 ASYNCcnt path), serve all
// neighbor reuse from LDS, stream outputs with non-temporal B128 stores.

typedef float v4f __attribute__((ext_vector_type(4)));

#define IMG_W     1024
#define IMG_H     1024
#define TILE_ROWS 8
#define THREADS   256   // 8 waves (wave32); 256*4 floats == one full row
#define NCH       3     // compile-time channel count: n/3, n%3 -> mul/shift

__global__ __launch_bounds__(THREADS) void grad_mix_kernel(
    const float* __restrict__ x, float* __restrict__ out) {
  // 9 rows: row 0 is the circular halo (r0-1 mod H), rows 1..8 are r0..r0+7.
  __shared__ float smem[(TILE_ROWS + 1) * IMG_W];

  const int t   = threadIdx.x;
  const int n   = blockIdx.y;               // image index (b*NCH + c), 0..47
  const int r0  = blockIdx.x * TILE_ROWS;   // first computed row of this tile
  const int b   = n / NCH;                  // constant div -> mul/shift
  const int c   = n % NCH;
  const int col = t * 4;                    // float column of this thread

  const size_t imgBase = (size_t)n * IMG_H * IMG_W;

  // --- Async stage TILE_ROWS+1 rows (global -> LDS), 1 x b128 per row per thread.
#pragma unroll
  for (int j = 0; j <= TILE_ROWS; ++j) {
    int rr = r0 - 1 + j;              // only j==0 can wrap
    if (rr < 0) rr += IMG_H;
    const float* gp = x + imgBase + (size_t)rr * IMG_W + col;
    unsigned lds_byte      = (unsigned)(size_t)(&smem[j * IMG_W + col]);
    unsigned long long gva = (unsigned long long)(size_t)gp;
    asm volatile("global_load_async_to_lds_b128 %0, %1, off"
                 :: "v"(lds_byte), "v"(gva)
                 : "memory");
  }
  asm volatile("s_wait_asynccnt 0" ::: "memory");   // drain this wave's ASYNCcnt
  __syncthreads();                                  // make all waves' rows visible

  // --- Compute 3 output channels per row, all operands served from LDS.
  const size_t chStride = (size_t)NCH * IMG_H * IMG_W;             // +3 channels
  const size_t outBase  = ((size_t)b * (3 * NCH) + c) * IMG_H * IMG_W; // x_d ch

#pragma unroll
  for (int lr = 0; lr < TILE_ROWS; ++lr) {
    const float* rowC = &smem[(lr + 1) * IMG_W];   // current row
    const float* rowU = &smem[lr * IMG_W];         // row above (incl. halo)

    float c0 = rowC[col + 0], c1 = rowC[col + 1];
    float c2 = rowC[col + 2], c3 = rowC[col + 3];
    float pw = rowC[(col + IMG_W - 1) & (IMG_W - 1)];   // circular w-1 neighbor
    float u0 = rowU[col + 0], u1 = rowU[col + 1];
    float u2 = rowU[col + 2], u3 = rowU[col + 3];

    v4f xd = {c0 - pw, c1 - c0, c2 - c1, c3 - c2};
    v4f yd = {c0 - u0, c1 - u1, c2 - u2, c3 - u3};
    v4f cp = {c0, c1, c2, c3};

    size_t o = outBase + (size_t)(r0 + lr) * IMG_W + col;
    // Write-once streaming outputs: non-temporal B128 stores keep L2 for input reuse.
    __builtin_nontemporal_store(xd, (v4f*)(out + o));
    __builtin_nontemporal_store(yd, (v4f*)(out + o + chStride));
    __builtin_nontemporal_store(cp, (v4f*)(out + o + 2 * chStride));
  }
}

extern "C" void kernel_launch(void* const* d_in, const int* in_sizes, int n_in,
                              void* d_out, int out_size, void* d_ws, size_t ws_size,
                              hipStream_t stream) {
  (void)n_in; (void)out_size; (void)d_ws; (void)ws_size;
  const float* x = (const float*)d_in[0];
  float* out     = (float*)d_out;

  const int nImages  = in_sizes[0] / (IMG_H * IMG_W);   // 48
  const int rowTiles = IMG_H / TILE_ROWS;               // 128

  hipLaunchKernelGGL(grad_mix_kernel, dim3(rowTiles, nImages), dim3(THREADS),
                     0, stream, x, out);
}